// GNN_10204842295505
// MI455X (gfx1250) — compile-verified
//
#include <hip/hip_runtime.h>

// ext vectors: b128 vmem ops + WMMA operand shapes (wave32)
typedef int   v4i __attribute__((ext_vector_type(4)));
typedef float v4f __attribute__((ext_vector_type(4)));
typedef float v2f __attribute__((ext_vector_type(2)));
typedef float v8f __attribute__((ext_vector_type(8)));

// Fire-and-forget f32 atomic add at device scope (L2 atomic units).
// No-return form -> tracked with STOREcnt only; s_endpgm's implicit wait-idle
// guarantees completion before the next kernel in the stream.
__device__ __forceinline__ void atomic_add_f32_noret(float* p, float v) {
    asm volatile("global_atomic_add_f32 %0, %1, off scope:SCOPE_DEV"
                 :: "v"(p), "v"(v)
                 : "memory");
}

// Random 4B gather served at device scope: skip WGP$ fill (no 128B line
// amplification for a 4MB table with ~zero L0 reuse), hit in L2.
__device__ __forceinline__ float gather_f32_dev(const float* p) {
    return __hip_atomic_load(p, __ATOMIC_RELAXED, __HIP_MEMORY_SCOPE_AGENT);
}

// ---------------------------------------------------------------------------
// Kernel 1: zero the neighbor-sum accumulator (4 MB in d_ws). Regular-temporal
// stores: this array must STAY resident in L2 for the atomic phase.
// ---------------------------------------------------------------------------
__global__ void gnn_zero_ws(float* __restrict__ ws, int n) {
    int base = (blockIdx.x * blockDim.x + threadIdx.x) << 2;
    if (base + 4 <= n) {
        *(v4f*)(ws + base) = (v4f){0.f, 0.f, 0.f, 0.f};
    } else if (base < n) {
        for (int i = base; i < n; ++i) ws[i] = 0.f;
    }
}

// ---------------------------------------------------------------------------
// Kernel 2: edge phase, 8 edges per lane. This is the HBM-roofline kernel
// (256 MB edge stream ~ 11 us at 23.3 TB/s).
//   - 4x b128 NT loads stream the edge lists (used once: keep them from
//     evicting the 8 MB hot set out of the 192 MB L2, no WGP$ pollution)
//   - 8 DEV-scope gathers of state[src] (compiler-tracked LOADcnt, 8-deep MLP)
//   - 8 no-return DEV-scope f32 atomics into ws[dst]
// ---------------------------------------------------------------------------
__global__ void gnn_edge_scatter(const float* __restrict__ state,
                                 const int*   __restrict__ esrc,
                                 const int*   __restrict__ edst,
                                 float*       __restrict__ ws,
                                 int ne) {
    int base = (blockIdx.x * blockDim.x + threadIdx.x) << 3;
    if (base + 8 <= ne) {
        v4i s_lo = __builtin_nontemporal_load((const v4i*)(esrc + base));
        v4i s_hi = __builtin_nontemporal_load((const v4i*)(esrc + base) + 1);
        v4i d_lo = __builtin_nontemporal_load((const v4i*)(edst + base));
        v4i d_hi = __builtin_nontemporal_load((const v4i*)(edst + base) + 1);
        // Issue all gathers first: 8 outstanding L2 loads per lane.
        float m0 = gather_f32_dev(state + s_lo.x);
        float m1 = gather_f32_dev(state + s_lo.y);
        float m2 = gather_f32_dev(state + s_lo.z);
        float m3 = gather_f32_dev(state + s_lo.w);
        float m4 = gather_f32_dev(state + s_hi.x);
        float m5 = gather_f32_dev(state + s_hi.y);
        float m6 = gather_f32_dev(state + s_hi.z);
        float m7 = gather_f32_dev(state + s_hi.w);
        atomic_add_f32_noret(ws + d_lo.x, m0);
        atomic_add_f32_noret(ws + d_lo.y, m1);
        atomic_add_f32_noret(ws + d_lo.z, m2);
        atomic_add_f32_noret(ws + d_lo.w, m3);
        atomic_add_f32_noret(ws + d_hi.x, m4);
        atomic_add_f32_noret(ws + d_hi.y, m5);
        atomic_add_f32_noret(ws + d_hi.z, m6);
        atomic_add_f32_noret(ws + d_hi.w, m7);
    } else if (base < ne) {
        for (int i = base; i < ne; ++i) {
            atomic_add_f32_noret(ws + edst[i], gather_f32_dev(state + esrc[i]));
        }
    }
}

// ---------------------------------------------------------------------------
// Kernel 3: node phase = the reference's actual matrix op,
//   out = [state, ns] @ W^T + b   ([n x 2] @ [2 x 1], K padded 2->4),
// mapped onto V_WMMA_F32_16X16X4_F32, one 16-node tile per wave.
//
//   A (16x4)  = W^T broadcast down rows:  (v0,lo)=K0=w0, (v1,lo)=K1=w1,
//               high half (K2,K3) = 0                      [documented layout]
//   B (4x16)  = node features per column: (v0,lo)=state,  (v1,lo)=ns,
//               high half (K2,K3) = 0
//   C (16x16) = bias broadcast.
// => D[m,n] = w0*state[n] + w1*ns[n] + b for every m (column-constant), so
// VGPR0 of lane l (l<16) is exactly out[base+l]: one predicated NT b32 store,
// no shuffles. Wave-uniform tile guard keeps EXEC all-1s around the WMMA.
// ---------------------------------------------------------------------------
__global__ void gnn_node_out_wmma(const float* __restrict__ state,
                                  const float* __restrict__ ws,
                                  const float* __restrict__ W,
                                  const float* __restrict__ b,
                                  float*       __restrict__ out,
                                  int n) {
    const int lane = threadIdx.x & 31;
    const int wave = blockIdx.x * (blockDim.x >> 5) + (threadIdx.x >> 5);
    const int base = wave << 4;                 // 16 nodes per wave
    const float w0 = W[0];
    const float w1 = W[1];
    const float bb = b[0];

    if (base + 16 <= n) {                       // wave-uniform: EXEC all-1s here
        float sv = 0.f, nv = 0.f;
        if (lane < 16) {                        // coalesced 64B loads (L2-hot)
            sv = state[base + lane];
            nv = ws[base + lane];
        }
        v2f amat, bmat;
        amat.x = (lane < 16) ? w0 : 0.f;        // A: K0 row (high half = K2 = 0)
        amat.y = (lane < 16) ? w1 : 0.f;        // A: K1 row (high half = K3 = 0)
        bmat.x = sv;                            // B: K0 row = state (0 in hi half)
        bmat.y = nv;                            // B: K1 row = ns    (0 in hi half)
        v8f c = {bb, bb, bb, bb, bb, bb, bb, bb};
        v8f d = __builtin_amdgcn_wmma_f32_16x16x4_f32(
            /*neg_a=*/false, amat, /*neg_b=*/false, bmat,
            /*c_mod=*/(short)0, c, /*reuse_a=*/false, /*reuse_b=*/false);
        if (lane < 16) {
            __builtin_nontemporal_store(d[0], out + base + lane);
        }
    } else if (base < n) {                      // boundary tile: scalar path
        int i = base + lane;
        if (lane < 16 && i < n) {
            out[i] = state[i] * w0 + ws[i] * w1 + bb;
        }
    }
}

// ---------------------------------------------------------------------------
// Launcher. Inputs (setup_inputs order):
//   d_in[0] state     float32 [N_NODES]
//   d_in[1] edge_src  int32   [N_EDGES]
//   d_in[2] edge_dst  int32   [N_EDGES]
//   d_in[3] W         float32 [2]
//   d_in[4] b         float32 [1]
// d_out: float32 [N_NODES]; d_ws: >= N_NODES*4 bytes accumulator.
// ---------------------------------------------------------------------------
extern "C" void kernel_launch(void* const* d_in, const int* in_sizes, int n_in,
                              void* d_out, int out_size, void* d_ws, size_t ws_size,
                              hipStream_t stream) {
    const float* state = (const float*)d_in[0];
    const int*   esrc  = (const int*)d_in[1];
    const int*   edst  = (const int*)d_in[2];
    const float* W     = (const float*)d_in[3];
    const float* b     = (const float*)d_in[4];
    float*       out   = (float*)d_out;
    float*       ws    = (float*)d_ws;

    const int n  = in_sizes[0];   // nodes
    const int ne = in_sizes[1];   // edges

    const int TB = 256;           // 8 wave32 waves per workgroup

    const int n4     = (n + 3) / 4;      // vec4 zeroing work-items
    const int ne8    = (ne + 7) / 8;     // 8-edge work-items
    const int ntiles = (n + 15) / 16;    // one 16-node WMMA tile per wave

    const int gZero = (n4 + TB - 1) / TB;
    const int gEdge = (ne8 + TB - 1) / TB;
    const int gNode = (ntiles + 7) / 8;  // 8 waves per 256-thread block

    gnn_zero_ws<<<gZero, TB, 0, stream>>>(ws, n);
    gnn_edge_scatter<<<gEdge, TB, 0, stream>>>(state, esrc, edst, ws, ne);
    gnn_node_out_wmma<<<gNode, TB, 0, stream>>>(state, ws, W, b, out, n);
}